// GNN_5781025981003
// MI455X (gfx1250) — compile-verified
//
#include <hip/hip_runtime.h>

// DMPNN forward for MI455X (gfx1250, wave32).
// fp32 WMMA (v_wmma_f32_16x16x4_f32) for all GEMMs — workload is HBM-bound
// (~3.5 GB traffic vs 57 GFLOP), so the fp32 matrix path is free and exact.
// LDS staging uses CDNA5 async-to-LDS (GLOBAL_LOAD_ASYNC_TO_LDS_B128 +
// s_wait_asynccnt) where the staging is a pure copy, overlapping the VALU
// gather work; guarded by __has_builtin with scalar fallbacks.
//
// Workspace layout (floats), requires ~794 MB:
//   h0   [E,H]  edge-init activations (residual source)
//   hA   [E,H]  ping
//   hB   [E,H]  pong
//   amsg [N,H]  scatter-add buffer (reused as `s` at the end)
//   pooled [G,H]

typedef float v2f __attribute__((ext_vector_type(2)));
typedef float v8f __attribute__((ext_vector_type(8)));
typedef int   v4i __attribute__((ext_vector_type(4)));

#define N_NODES 50000
#define N_EDGES 500000
#define HDIM    128
#define FN      32
#define FE      16
#define NGRAPH  128
#define GNN_DEPTH 3

#if defined(__has_builtin)
#  if __has_builtin(__builtin_amdgcn_global_load_async_to_lds_b128)
#    define HAVE_ASYNC_LDS 1
#  endif
#  if __has_builtin(__builtin_amdgcn_s_wait_asynccnt)
#    define HAVE_WAIT_ASYNC 1
#  endif
#endif

#define AS1 __attribute__((address_space(1)))
#define AS3 __attribute__((address_space(3)))

__device__ __forceinline__ void async_copy16(const float* g, float* l) {
#if defined(HAVE_ASYNC_LDS)
    __builtin_amdgcn_global_load_async_to_lds_b128(
        (AS1 v4i*)g, (AS3 v4i*)l, 0, 0);
#else
    const float4 v = *(const float4*)g;
    *(float4*)l = v;
#endif
}

__device__ __forceinline__ void async_wait0() {
#if defined(HAVE_WAIT_ASYNC)
    __builtin_amdgcn_s_wait_asynccnt(0);
#elif defined(HAVE_ASYNC_LDS)
    asm volatile("s_wait_asynccnt 0x0" ::: "memory");
#endif
}

__device__ __forceinline__ int imin_d(int a, int b) { return a < b ? a : b; }

// ---------------------------------------------------------------------------
// zero-fill
// ---------------------------------------------------------------------------
__global__ void zero_kernel(float* __restrict__ p, int n) {
    int i = blockIdx.x * blockDim.x + threadIdx.x;
    if (i < n) p[i] = 0.0f;
}

// ---------------------------------------------------------------------------
// edge init: h0[e,:] = relu( cat(x[row[e]], edge_attr[e]) @ W_init + b_init )
// GEMM [E,48] x [48,128]. One block = 128 edges = 8 waves x 16-edge M-tiles.
// Feature gather is a pure copy -> async-to-LDS b128 per 4-float chunk.
// ---------------------------------------------------------------------------
__global__ __launch_bounds__(256)
void edge_init_kernel(const float* __restrict__ x,
                      const float* __restrict__ ea,
                      const int*   __restrict__ ei,     // row = ei[e]
                      const float* __restrict__ Wi,     // [48][128]
                      const float* __restrict__ bi,     // [128]
                      float* __restrict__ h0)
{
    __shared__ __align__(16) float Bs[48 * HDIM];   // full W_init: 24.5 KB
    __shared__ __align__(16) float As[8][16 * 48];  // per-wave A tiles: 24.5 KB

    const int tid  = threadIdx.x;
    const int w    = tid >> 5;
    const int lane = tid & 31;
    const int e0   = blockIdx.x * 128 + w * 16;

    // stage W_init (contiguous): b128 async copies, 6144 floats / block
    for (int i = tid; i < (48 * HDIM) / 4; i += 256)
        async_copy16(Wi + i * 4, &Bs[i * 4]);

    // gather x[row[e]] (8 chunks) per edge
    for (int i = lane; i < 16 * 8; i += 32) {        // uniform trip count
        int m = i >> 3, c4 = i & 7;
        int e = imin_d(e0 + m, N_EDGES - 1);
        async_copy16(x + ei[e] * FN + c4 * 4, &As[w][m * 48 + c4 * 4]);
    }
    // gather edge_attr[e] (4 chunks) per edge
    for (int i = lane; i < 16 * 4; i += 32) {
        int m = i >> 2, c4 = i & 3;
        int e = imin_d(e0 + m, N_EDGES - 1);
        async_copy16(ea + e * FE + c4 * 4, &As[w][m * 48 + FN + c4 * 4]);
    }
    async_wait0();
    __syncthreads();

    const int nl   = lane & 15;
    const int kl   = lane >> 4;
    const int koff = kl * 2;
    const int m    = nl;

    v8f acc[8];
    #pragma unroll
    for (int j = 0; j < 8; ++j) {
        float bv = bi[j * 16 + nl];
        #pragma unroll
        for (int i = 0; i < 8; ++i) acc[j][i] = bv;
    }

    #pragma unroll
    for (int k0 = 0; k0 < 48; k0 += 4) {
        v2f a = *(const v2f*)&As[w][m * 48 + k0 + koff];
        #pragma unroll
        for (int j = 0; j < 8; ++j) {
            v2f b;
            b.x = Bs[(k0 + koff)     * HDIM + j * 16 + nl];
            b.y = Bs[(k0 + koff + 1) * HDIM + j * 16 + nl];
            acc[j] = __builtin_amdgcn_wmma_f32_16x16x4_f32(
                false, a, false, b, (short)0, acc[j], false, false);
        }
    }

    #pragma unroll
    for (int i = 0; i < 8; ++i) {
        int e = e0 + i + kl * 8;
        if (e < N_EDGES) {
            #pragma unroll
            for (int j = 0; j < 8; ++j) {
                float v = acc[j][i];
                h0[e * HDIM + j * 16 + nl] = v > 0.0f ? v : 0.0f;
            }
        }
    }
}

// ---------------------------------------------------------------------------
// scatter-add: out[col[e], f] += h[e, f]   (segment_sum over edge targets)
// one thread = 4 consecutive features of one edge
// ---------------------------------------------------------------------------
__global__ void scatter_add_kernel(const float* __restrict__ h,
                                   const int*   __restrict__ col,
                                   float* __restrict__ out)
{
    int t = blockIdx.x * blockDim.x + threadIdx.x;     // E*32 = 16M threads
    if (t >= N_EDGES * 32) return;
    int e  = t >> 5;
    int fq = (t & 31) * 4;
    int c  = col[e];
    const float4 v = *(const float4*)(h + (size_t)e * HDIM + fq);
    float* dst = out + (size_t)c * HDIM + fq;
    atomicAdd(dst + 0, v.x);
    atomicAdd(dst + 1, v.y);
    atomicAdd(dst + 2, v.z);
    atomicAdd(dst + 3, v.w);
}

// ---------------------------------------------------------------------------
// conv layer: hnext[e,:] = relu( (amsg[row[e],:] - hcur[e^1,:]) @ W + b + h0[e,:] )
// GEMM [E,128] x [128,128], K chunked by 32 through LDS.
// Weight chunk staged with async-to-LDS, overlapping the VALU A-tile gather.
// ---------------------------------------------------------------------------
__global__ __launch_bounds__(256)
void conv_kernel(const float* __restrict__ hcur,
                 const float* __restrict__ h0,
                 const float* __restrict__ amsg,
                 const int*   __restrict__ ei,      // row = ei[e]
                 const float* __restrict__ W,       // [128][128]
                 const float* __restrict__ b,       // [128]
                 float* __restrict__ hnext)
{
    __shared__ __align__(16) float Ws[32 * HDIM];   // 16 KB weight K-chunk
    __shared__ __align__(16) float As[8][16 * 32];  // 16 KB per-wave A chunks

    const int tid  = threadIdx.x;
    const int w    = tid >> 5;
    const int lane = tid & 31;
    const int e0   = blockIdx.x * 128 + w * 16;

    const int nl   = lane & 15;
    const int kl   = lane >> 4;
    const int koff = kl * 2;
    const int m    = nl;

    v8f acc[8];
    #pragma unroll
    for (int j = 0; j < 8; ++j) {
        float bv = b[j * 16 + nl];
        #pragma unroll
        for (int i = 0; i < 8; ++i) acc[j][i] = bv;
    }

    for (int kc = 0; kc < 4; ++kc) {
        __syncthreads();
        // async weight chunk copy (contiguous 16 KB), overlaps gather below
        for (int i = tid; i < (32 * HDIM) / 4; i += 256)
            async_copy16(W + kc * 32 * HDIM + i * 4, &Ws[i * 4]);
        // VALU gather: message = amsg[row[e]] - h[e^1]  (pair-reverse)
        for (int i = lane; i < 16 * 32; i += 32) {         // uniform trip count
            int mm = i >> 5, k = i & 31;
            int e  = imin_d(e0 + mm, N_EDGES - 1);
            int r  = ei[e];
            As[w][i] = amsg[r * HDIM + kc * 32 + k]
                     - hcur[(e ^ 1) * HDIM + kc * 32 + k];
        }
        if (kc < 3)   // warm next weight chunk: global_prefetch_b8
            __builtin_prefetch(&W[(kc + 1) * 32 * HDIM + tid * 16], 0, 1);
        async_wait0();
        __syncthreads();

        #pragma unroll
        for (int k0 = 0; k0 < 32; k0 += 4) {
            v2f a = *(const v2f*)&As[w][m * 32 + k0 + koff];
            #pragma unroll
            for (int j = 0; j < 8; ++j) {
                v2f bb;
                bb.x = Ws[(k0 + koff)     * HDIM + j * 16 + nl];
                bb.y = Ws[(k0 + koff + 1) * HDIM + j * 16 + nl];
                acc[j] = __builtin_amdgcn_wmma_f32_16x16x4_f32(
                    false, a, false, bb, (short)0, acc[j], false, false);
            }
        }
    }

    #pragma unroll
    for (int i = 0; i < 8; ++i) {
        int e = e0 + i + kl * 8;
        if (e < N_EDGES) {
            #pragma unroll
            for (int j = 0; j < 8; ++j) {
                int n   = j * 16 + nl;
                float v = acc[j][i] + h0[e * HDIM + n];
                hnext[e * HDIM + n] = v > 0.0f ? v : 0.0f;
            }
        }
    }
}

// ---------------------------------------------------------------------------
// node head: hn = relu( cat(x, s) @ W_e2n + b ); pooled[batch[v]] += hn (atomic)
// GEMM [N,160] x [160,128], K chunked by 32. Both weight chunk and A gather
// are pure copies -> fully async-to-LDS.
// ---------------------------------------------------------------------------
__global__ __launch_bounds__(256)
void node_kernel(const float* __restrict__ x,
                 const float* __restrict__ s,
                 const int*   __restrict__ batch,
                 const float* __restrict__ W,       // [160][128]
                 const float* __restrict__ b,       // [128]
                 float* __restrict__ pooled)        // [G][128]
{
    __shared__ __align__(16) float Ws[32 * HDIM];
    __shared__ __align__(16) float As[8][16 * 32];

    const int tid  = threadIdx.x;
    const int w    = tid >> 5;
    const int lane = tid & 31;
    const int v0   = blockIdx.x * 128 + w * 16;

    const int nl   = lane & 15;
    const int kl   = lane >> 4;
    const int koff = kl * 2;
    const int m    = nl;

    v8f acc[8];
    #pragma unroll
    for (int j = 0; j < 8; ++j) {
        float bv = b[j * 16 + nl];
        #pragma unroll
        for (int i = 0; i < 8; ++i) acc[j][i] = bv;
    }

    for (int kc = 0; kc < 5; ++kc) {
        __syncthreads();
        for (int i = tid; i < (32 * HDIM) / 4; i += 256)
            async_copy16(W + kc * 32 * HDIM + i * 4, &Ws[i * 4]);
        // A chunk: cat(x, s) rows, 4-float chunks (kc*32 aligned so each
        // chunk lies fully inside either the x part or the s part)
        for (int i = lane; i < 16 * 8; i += 32) {
            int mm = i >> 3, c4 = i & 7;
            int vv = imin_d(v0 + mm, N_NODES - 1);
            int kg = kc * 32 + c4 * 4;
            const float* src = (kg < FN) ? (x + vv * FN + kg)
                                         : (s + vv * HDIM + (kg - FN));
            async_copy16(src, &As[w][mm * 32 + c4 * 4]);
        }
        if (kc < 4)
            __builtin_prefetch(&W[(kc + 1) * 32 * HDIM + tid * 16], 0, 1);
        async_wait0();
        __syncthreads();

        #pragma unroll
        for (int k0 = 0; k0 < 32; k0 += 4) {
            v2f a = *(const v2f*)&As[w][m * 32 + k0 + koff];
            #pragma unroll
            for (int j = 0; j < 8; ++j) {
                v2f bb;
                bb.x = Ws[(k0 + koff)     * HDIM + j * 16 + nl];
                bb.y = Ws[(k0 + koff + 1) * HDIM + j * 16 + nl];
                acc[j] = __builtin_amdgcn_wmma_f32_16x16x4_f32(
                    false, a, false, bb, (short)0, acc[j], false, false);
            }
        }
    }

    #pragma unroll
    for (int i = 0; i < 8; ++i) {
        int v = v0 + i + kl * 8;
        if (v < N_NODES) {
            int g = batch[v];
            #pragma unroll
            for (int j = 0; j < 8; ++j) {
                float val = acc[j][i];
                val = val > 0.0f ? val : 0.0f;
                atomicAdd(&pooled[g * HDIM + j * 16 + nl], val);
            }
        }
    }
}

// ---------------------------------------------------------------------------
// final FFN: out[g] = pooled[g,:] . W_ffn + b_ffn     (tiny; one block)
// ---------------------------------------------------------------------------
__global__ void ffn_kernel(const float* __restrict__ pooled,
                           const float* __restrict__ Wf,   // [128]
                           const float* __restrict__ bf,   // [1]
                           float* __restrict__ out)
{
    int g = threadIdx.x;
    if (g >= NGRAPH) return;
    float acc = bf[0];
    #pragma unroll 8
    for (int k = 0; k < HDIM; ++k) acc += pooled[g * HDIM + k] * Wf[k];
    out[g] = acc;
}

// ---------------------------------------------------------------------------
extern "C" void kernel_launch(void* const* d_in, const int* in_sizes, int n_in,
                              void* d_out, int out_size, void* d_ws, size_t ws_size,
                              hipStream_t stream)
{
    (void)in_sizes; (void)n_in; (void)out_size; (void)ws_size;

    const float* x       = (const float*)d_in[0];
    const float* ea      = (const float*)d_in[1];
    const int*   ei      = (const int*)  d_in[2];   // [2][E]: row then col
    const int*   batch   = (const int*)  d_in[3];
    const float* W_init  = (const float*)d_in[4];
    const float* b_init  = (const float*)d_in[5];
    const float* W_convs = (const float*)d_in[6];   // [3][128][128]
    const float* b_convs = (const float*)d_in[7];   // [3][128]
    const float* W_e2n   = (const float*)d_in[8];   // [160][128]
    const float* b_e2n   = (const float*)d_in[9];
    const float* W_ffn   = (const float*)d_in[10];  // [128]
    const float* b_ffn   = (const float*)d_in[11];
    float*       out     = (float*)d_out;

    float* ws     = (float*)d_ws;
    float* h0     = ws;
    float* hA     = h0 + (size_t)N_EDGES * HDIM;
    float* hB     = hA + (size_t)N_EDGES * HDIM;
    float* amsg   = hB + (size_t)N_EDGES * HDIM;
    float* pooled = amsg + (size_t)N_NODES * HDIM;

    const int* col = ei + N_EDGES;

    const dim3 blk(256);
    const int egrid  = (N_EDGES + 127) / 128;        // 3907
    const int ngrid  = (N_NODES + 127) / 128;        // 391
    const int zgridA = (N_NODES * HDIM + 255) / 256;
    const int zgridP = (NGRAPH * HDIM + 255) / 256;
    const int sgrid  = (N_EDGES * 32 + 255) / 256;   // 62500

    edge_init_kernel<<<egrid, blk, 0, stream>>>(x, ea, ei, W_init, b_init, h0);

    const float* hc = h0;
    float*       hn = hA;
    for (int l = 0; l < GNN_DEPTH; ++l) {
        zero_kernel<<<zgridA, blk, 0, stream>>>(amsg, N_NODES * HDIM);
        scatter_add_kernel<<<sgrid, blk, 0, stream>>>(hc, col, amsg);
        conv_kernel<<<egrid, blk, 0, stream>>>(hc, h0, amsg, ei,
                                               W_convs + (size_t)l * HDIM * HDIM,
                                               b_convs + (size_t)l * HDIM, hn);
        hc = hn;
        hn = (hn == hA) ? hB : hA;
    }

    // s = segment_sum(h, col)  (reuse amsg buffer)
    zero_kernel<<<zgridA, blk, 0, stream>>>(amsg, N_NODES * HDIM);
    scatter_add_kernel<<<sgrid, blk, 0, stream>>>(hc, col, amsg);

    zero_kernel<<<zgridP, blk, 0, stream>>>(pooled, NGRAPH * HDIM);
    node_kernel<<<ngrid, blk, 0, stream>>>(x, amsg, batch, W_e2n, b_e2n, pooled);

    ffn_kernel<<<1, 128, 0, stream>>>(pooled, W_ffn, b_ffn, out);
}